// GPTNeoXAttention_63505386438932
// MI455X (gfx1250) — compile-verified
//
#include <hip/hip_runtime.h>
#include <cstdint>
#include <cstddef>

// Problem constants (fixed by the reference): B=2, S=2048, H=32, D=128, P=2048
#define B_   2
#define S_   2048
#define H_   32
#define D_   128
#define P_   2048
#define T_   (P_ + S_)     // 4096 total cache rows
#define ROT_ 32

typedef float v4f __attribute__((ext_vector_type(4)));
typedef int   v4i __attribute__((ext_vector_type(4)));

// -log2(10000)/16 : inv_freq[fi] = 10000^(-fi/16) = exp2(fi * NEG_L2B_OVER_16)
#define NEG_L2B_OVER_16 (-0.8304820237218406f)

#ifndef __has_builtin
#define __has_builtin(x) 0
#endif
#if __has_builtin(__builtin_amdgcn_global_load_async_to_lds_b128) && \
    __has_builtin(__builtin_amdgcn_global_store_async_from_lds_b128) && \
    __has_builtin(__builtin_amdgcn_s_wait_asynccnt)
#define USE_ASYNC_LDS 1
#else
#define USE_ASYNC_LDS 0
#endif

// ---------------------------------------------------------------------------
// Kernel 0: RoPE cos/sin table, full libm accuracy, computed once per launch.
// Layout: tbl[s*32 + fi] = cos(angle), tbl[s*32 + 16 + fi] = sin(angle),
// angle = (P + s) * 10000^(-fi/16).  Total 2048*32 floats = 256 KB.
// ---------------------------------------------------------------------------
__global__ __launch_bounds__(256) void rope_table(float* __restrict__ tbl) {
  const int i  = blockIdx.x * 256 + threadIdx.x;   // [0, S*16)
  const int s  = i >> 4;
  const int fi = i & 15;
  const float inv = exp2f((float)fi * NEG_L2B_OVER_16);
  float sn, cs;
  sincosf((float)(P_ + s) * inv, &sn, &cs);
  tbl[s * 32 + fi]      = cs;
  tbl[s * 32 + 16 + fi] = sn;
}

// ---------------------------------------------------------------------------
// Kernel 1: fused QKV split + transpose + RoPE.
// One wave32 per (b, s, h) row; each lane owns one float4 (4 dims) of each of
// the q/k/v slices.  RoPE partner x[d^16] lives in lane^4, same component.
// USE_TBL=true reads cos/sin from the (hot, L2-resident) table instead of
// recomputing transcendental per wave.
// ---------------------------------------------------------------------------
template <bool USE_TBL>
__global__ __launch_bounds__(256) void qkv_rope_split(
    const float* __restrict__ qkv,
    const float* __restrict__ tbl,
    float* __restrict__ qo,
    float* __restrict__ ko,
    float* __restrict__ vo) {
  const int lane = threadIdx.x & 31;
  const int wid  = blockIdx.x * 8 + (threadIdx.x >> 5);   // row = ((b*S+s)*H+h)
  const int h  = wid & (H_ - 1);
  const int bs = wid >> 5;            // b*S + s   (H==32)
  const int s  = bs & (S_ - 1);
  const int b  = bs >> 11;            // S==2048

  // input row base: wid * 3*D floats; q at +0, k at +D, v at +2D
  const v4f* in4 = (const v4f*)(qkv + (size_t)wid * (3 * D_));
  v4f xq = __builtin_nontemporal_load(in4 + lane);
  v4f xk = __builtin_nontemporal_load(in4 + 32 + lane);
  v4f xv = __builtin_nontemporal_load(in4 + 64 + lane);

  const int d0 = lane * 4;
  if (d0 < ROT_) {                    // lanes 0..7 carry the rotated dims
    // partner values x[d ^ 16] sit in lane ^ 4, same vector component
    v4f pq, pk;
    pq.x = __shfl_xor(xq.x, 4, 32);
    pq.y = __shfl_xor(xq.y, 4, 32);
    pq.z = __shfl_xor(xq.z, 4, 32);
    pq.w = __shfl_xor(xq.w, 4, 32);
    pk.x = __shfl_xor(xk.x, 4, 32);
    pk.y = __shfl_xor(xk.y, 4, 32);
    pk.z = __shfl_xor(xk.z, 4, 32);
    pk.w = __shfl_xor(xk.w, 4, 32);

    const float sign = (d0 < 16) ? -1.0f : 1.0f;   // rot = concat(-x2, x1)
    v4f c, sn;
    if (USE_TBL) {
      // 8 float4 per position: [0..3]=cos, [4..7]=sin.  Temporal loads —
      // 256 KB table reused 64x across (b, h); keep it cached.
      const v4f* t4 = (const v4f*)(tbl + (size_t)s * 32);
      c  = t4[lane & 3];
      sn = t4[4 + (lane & 3)];
    } else {
      const float pos = (float)(P_ + s);
      const int   fb  = d0 & 15;                   // freq index base
      float cv[4], sv[4];
#pragma unroll
      for (int j = 0; j < 4; ++j) {
        const float inv = exp2f((float)(fb + j) * NEG_L2B_OVER_16);
        sincosf(pos * inv, &sv[j], &cv[j]);
      }
      c  = (v4f){cv[0], cv[1], cv[2], cv[3]};
      sn = (v4f){sv[0], sv[1], sv[2], sv[3]};
    }
    xq = xq * c + (sign * pq) * sn;
    xk = xk * c + (sign * pk) * sn;
  }

  // q: (B,H,S,D) row-major
  const size_t qIdx = (((size_t)b * H_ + h) * S_ + s) * (D_ / 4) + lane;
  __builtin_nontemporal_store(xq, (v4f*)qo + qIdx);

  // k, v: (B,H,T,D); new tokens occupy rows [P, T)
  const size_t kvIdx =
      (((size_t)b * H_ + h) * T_ + (P_ + s)) * (D_ / 4) + lane;
  __builtin_nontemporal_store(xk, (v4f*)ko + kvIdx);
  __builtin_nontemporal_store(xv, (v4f*)vo + kvIdx);
}

// ---------------------------------------------------------------------------
// Kernel 2: past_key/past_value -> front of each (b,h) slab of k/v.
// blockIdx.y selects key (0) or value (1).  src slab stride = P*D/4 = 65536
// float4 (1<<16); dst slab stride = T*D/4 = 131072 float4 (1<<17).  Uses the
// CDNA5 async global->LDS->global stream (ASYNCcnt) when available.
// ---------------------------------------------------------------------------
#if USE_ASYNC_LDS
typedef __attribute__((address_space(1))) v4i gv4i;   // global int4
typedef __attribute__((address_space(3))) v4i sv4i;   // LDS int4
#endif

__global__ __launch_bounds__(256) void cache_copy(
    const float* __restrict__ k_src, const float* __restrict__ v_src,
    float* __restrict__ k_dst, float* __restrict__ v_dst) {
  const float* src = blockIdx.y ? v_src : k_src;
  float*       dst = blockIdx.y ? v_dst : k_dst;

  const unsigned idx    = blockIdx.x * 256u + threadIdx.x;  // float4 index
  const unsigned slab   = idx >> 16;
  const unsigned within = idx & 0xFFFFu;
  const size_t srcOff = (size_t)idx * 4;                           // floats
  const size_t dstOff = (((size_t)slab << 17) + (size_t)within) * 4;

#if USE_ASYNC_LDS
  __shared__ __align__(16) float lbuf[256 * 4];
  v4i* src_g = (v4i*)(const_cast<float*>(src) + srcOff);  // generic bitcast
  v4i* dst_g = (v4i*)(dst + dstOff);
  v4i* lds_g = (v4i*)(&lbuf[threadIdx.x * 4]);
  __builtin_amdgcn_global_load_async_to_lds_b128((gv4i*)src_g, (sv4i*)lds_g,
                                                 0, 0);
  __builtin_amdgcn_s_wait_asynccnt(0);
  __builtin_amdgcn_global_store_async_from_lds_b128((gv4i*)dst_g, (sv4i*)lds_g,
                                                    0, 0);
  // S_ENDPGM performs an implicit wait-idle; no trailing wait needed.
#else
  v4f val = __builtin_nontemporal_load((const v4f*)(src + srcOff));
  __builtin_nontemporal_store(val, (v4f*)(dst + dstOff));
#endif
}

// ---------------------------------------------------------------------------
extern "C" void kernel_launch(void* const* d_in, const int* in_sizes, int n_in,
                              void* d_out, int out_size, void* d_ws,
                              size_t ws_size, hipStream_t stream) {
  (void)in_sizes; (void)n_in; (void)out_size;

  const float* qkv   = (const float*)d_in[0];
  const float* pastK = (const float*)d_in[1];
  const float* pastV = (const float*)d_in[2];

  float* q = (float*)d_out;
  float* k = q + (size_t)B_ * H_ * S_ * D_;   // +16,777,216
  float* v = k + (size_t)B_ * H_ * T_ * D_;   // +33,554,432

  const size_t tblBytes = (size_t)S_ * 32 * sizeof(float);   // 256 KB
  const bool useTbl = (d_ws != nullptr) && (ws_size >= tblBytes);

  if (useTbl) {
    // 32768 (s, fi) pairs, one thread each
    rope_table<<<(S_ * 16) / 256, 256, 0, stream>>>((float*)d_ws);
    qkv_rope_split<true><<<(B_ * S_ * H_) / 8, 256, 0, stream>>>(
        qkv, (const float*)d_ws, q, k, v);
  } else {
    qkv_rope_split<false><<<(B_ * S_ * H_) / 8, 256, 0, stream>>>(
        qkv, nullptr, q, k, v);
  }

  // 4,194,304 float4 per cache tensor, 1 per thread; y: 0=key, 1=value
  const int n4 = (B_ * H_ * P_ * D_) / 4;
  cache_copy<<<dim3(n4 / 256, 2), 256, 0, stream>>>(pastK, pastV, k, v);
}